// Pool_66529043415276
// MI455X (gfx1250) — compile-verified
//
#include <hip/hip_runtime.h>
#include <stdint.h>

#define N_NODES 12288
#define D_FEAT  128
#define K_SEL   6144
#define M_SORT  16384   // next pow2 >= N_NODES, padded with key=0

typedef __attribute__((ext_vector_type(2))) float v2f;
typedef __attribute__((ext_vector_type(8))) float v8f;
typedef __attribute__((ext_vector_type(4))) unsigned int v4u;
typedef __attribute__((ext_vector_type(4))) int v4i;
typedef __attribute__((ext_vector_type(8))) int v8i;

// ---------------------------------------------------------------------------
// Kernel 1: scores = sigmoid(h @ W + b), packed into sortable 64-bit keys.
// One wave computes 16 rows via V_WMMA_F32_16X16X4_F32 (full f32 precision),
// looping K = 128 in chunks of 4. B carries W in column 0 only, so D[:,0]
// accumulates the dot products. Key = (bits(score)<<32) | ~row -> descending
// sort gives (score desc, row asc on ties), matching jax.lax.top_k.
// ---------------------------------------------------------------------------
__global__ __launch_bounds__(256) void score_keys_kernel(
    const float* __restrict__ h, const float* __restrict__ W,
    const float* __restrict__ b, unsigned long long* __restrict__ keys)
{
    const int lane = threadIdx.x & 31;
    const int wave = threadIdx.x >> 5;
    const int tile = blockIdx.x * 8 + wave;     // 96 blocks * 8 waves = 768 tiles
    const int m0   = tile * 16;
    const int half = lane >> 4;                 // 0: K pair {0,1}, 1: {2,3}
    const int sub  = lane & 15;

    const float* __restrict__ hrow = h + (size_t)(m0 + sub) * D_FEAT;

    v8f c = {};
    #pragma unroll 4
    for (int kk = 0; kk < 32; ++kk) {
        const int k0 = kk * 4 + half * 2;
        v2f a;
        a.x = hrow[k0];
        a.y = hrow[k0 + 1];
        v2f bb;
        const float w0 = W[k0];
        const float w1 = W[k0 + 1];
        bb.x = (sub == 0) ? w0 : 0.0f;          // only column N=0 carries W
        bb.y = (sub == 0) ? w1 : 0.0f;
        c = __builtin_amdgcn_wmma_f32_16x16x4_f32(
                false, a, false, bb, (short)0, c, false, false);
    }

    // D layout: column N=0 lives in lanes 0 and 16; VGPR v holds M = v + 8*half.
    if (sub == 0) {
        const float bias = b[0];
        #pragma unroll
        for (int v = 0; v < 8; ++v) {
            const int r = m0 + half * 8 + v;
            const float x = c[v] + bias;
            const float s = 1.0f / (1.0f + expf(-x));
            keys[r] = ((unsigned long long)__float_as_uint(s) << 32)
                      | (unsigned int)(~r);
        }
    }

    // Pad entries N_NODES..M_SORT-1 with key 0 (sorts to the end, descending).
    const int gt = blockIdx.x * 256 + threadIdx.x;
    if (gt < (M_SORT - N_NODES)) keys[N_NODES + gt] = 0ULL;
}

// ---------------------------------------------------------------------------
// Kernel 2: single-workgroup bitonic sort of 16384 u64 keys in 128 KB LDS
// (CDNA5: up to 320 KB LDS per workgroup). Descending order.
// Emits idx (int + float form) and values for the top K_SEL.
// ---------------------------------------------------------------------------
__global__ __launch_bounds__(1024) void sort_topk_kernel(
    const unsigned long long* __restrict__ keys_in,
    int* __restrict__ idx_i32, float* __restrict__ values,
    float* __restrict__ out_idx)
{
    __shared__ unsigned long long sk[M_SORT];   // 131072 bytes of LDS
    const int tid = threadIdx.x;

    for (int i = tid; i < M_SORT; i += 1024) sk[i] = keys_in[i];
    __syncthreads();

    for (int k = 2; k <= M_SORT; k <<= 1) {
        for (int j = k >> 1; j > 0; j >>= 1) {
            for (int t = tid; t < (M_SORT / 2); t += 1024) {
                const int i = ((t & ~(j - 1)) << 1) | (t & (j - 1));
                const int l = i | j;
                const bool desc = ((i & k) == 0);
                const unsigned long long a = sk[i];
                const unsigned long long bb = sk[l];
                const bool sw = desc ? (a < bb) : (a > bb);
                if (sw) { sk[i] = bb; sk[l] = a; }
            }
            __syncthreads();
        }
    }

    for (int i = tid; i < K_SEL; i += 1024) {
        const unsigned long long key = sk[i];
        const unsigned int sbits = (unsigned int)(key >> 32);
        const int id = (int)(~(unsigned int)(key & 0xFFFFFFFFu));
        idx_i32[i] = id;
        values[i]  = __uint_as_float(sbits);
        out_idx[i] = (float)id;
    }
}

// ---------------------------------------------------------------------------
// Kernel 3: new_h[i,:] = h[idx[i],:] * values[i]   (float4 vectorized)
// ---------------------------------------------------------------------------
__global__ __launch_bounds__(256) void newh_kernel(
    const float* __restrict__ h, const int* __restrict__ idx_i32,
    const float* __restrict__ values, float* __restrict__ out_newh)
{
    const int gid = blockIdx.x * 256 + threadIdx.x;   // 196608 float4 slots
    const int row = gid >> 5;                         // 32 float4 per row
    const int seg = gid & 31;
    const int src = idx_i32[row];
    const float v = values[row];
    const float4 hv = *(const float4*)(h + (size_t)src * D_FEAT + seg * 4);
    float4 o;
    o.x = hv.x * v; o.y = hv.y * v; o.z = hv.z * v; o.w = hv.w * v;
    *(float4*)(out_newh + (size_t)gid * 4) = o;
}

// ---------------------------------------------------------------------------
// Kernel 4 (the bandwidth eater): sub_g[i][j] = (g[r][c]!=0) + (r==c),
// r = idx[i], c = idx[j]. The 24 KB column-index table is staged into LDS by
// the Tensor Data Mover (one DMA descriptor, TENSORcnt-tracked), keeping the
// vector-memory pipe free for the scattered g gathers. g is streamed with
// non-temporal loads; output written with non-temporal stores.
// ---------------------------------------------------------------------------
__global__ __launch_bounds__(256) void subg_kernel(
    const float* __restrict__ g, const int* __restrict__ idx_i32,
    float* __restrict__ out_sub)
{
    __shared__ int sidx[K_SEL];                 // 24 KB of LDS

    if (threadIdx.x < 32) {
        // Flat pointer to LDS: aperture => addr[31:0] is the LDS byte offset.
        const unsigned int lds_off = (unsigned int)(uintptr_t)(void*)sidx;
        const unsigned long long ga = (unsigned long long)(uintptr_t)idx_i32;

        // D# group 0: count=1 | lds_addr | global_addr[120:64] | type=2
        v4u g0;
        g0.x = 1u;                                   // count=1 (valid), no gather
        g0.y = lds_off;                              // lds_addr
        g0.z = (unsigned int)(ga & 0xFFFFFFFFull);   // global_addr[95:64]
        g0.w = (unsigned int)(ga >> 32) | (2u << 30);// global_addr[120:96] | type=2

        // D# group 1: data_size=4B; 2-D tensor 6144 x 1; tile 6144 x 1.
        v8i g1;
        g1[0] = 0x00020000;          // workgroup_mask=0, data_size=2 (4 bytes)
        g1[1] = 0x18000000;          // tensor_dim0[15:0]=6144 in bits [63:48]
        g1[2] = 0x00010000;          // tensor_dim1[15:0]=1   in bits [95:80]
        g1[3] = 0x18000000;          // tile_dim0=6144        in bits [127:112]
        g1[4] = 0x00000001;          // tile_dim1=1           in bits [143:128]
        g1[5] = 6144;                // tensor_dim0_stride    in bits [191:160]
        g1[6] = 0;
        g1[7] = 0;

        v4i gz4 = {};                // groups 2/3 unused (<=2-D tensor)
        v8i gz8 = {};                // extra group (6-arg toolchain form)
        __builtin_amdgcn_tensor_load_to_lds(g0, g1, gz4, gz4, gz8, 0);
        __builtin_amdgcn_s_wait_tensorcnt(0);
    }
    __syncthreads();

    const int i = blockIdx.x;
    const int r = sidx[i];
    const float* __restrict__ grow = g + (size_t)r * N_NODES;
    float* __restrict__ orow = out_sub + (size_t)i * K_SEL;

    #pragma unroll 4
    for (int j = threadIdx.x; j < K_SEL; j += 256) {
        const int c = sidx[j];
        const float gv = __builtin_nontemporal_load(grow + c);
        float val = (gv != 0.0f) ? 1.0f : 0.0f;
        if (r == c) val += 1.0f;
        __builtin_nontemporal_store(val, orow + j);
    }
}

// ---------------------------------------------------------------------------
extern "C" void kernel_launch(void* const* d_in, const int* in_sizes, int n_in,
                              void* d_out, int out_size, void* d_ws, size_t ws_size,
                              hipStream_t stream)
{
    const float* g = (const float*)d_in[0];   // [N,N]
    const float* h = (const float*)d_in[1];   // [N,D]
    const float* W = (const float*)d_in[2];   // [D,1]
    const float* b = (const float*)d_in[3];   // [1]

    float* out = (float*)d_out;
    float* out_sub  = out;                                              // [K,K]
    float* out_newh = out + (size_t)K_SEL * K_SEL;                      // [K,D]
    float* out_idx  = out + (size_t)K_SEL * K_SEL + (size_t)K_SEL * D_FEAT; // [K]

    char* ws = (char*)d_ws;
    unsigned long long* keys = (unsigned long long*)ws;                 // 128 KB
    int*   idx_i32 = (int*)  (ws + (size_t)M_SORT * 8);                 // 24 KB
    float* values  = (float*)(ws + (size_t)M_SORT * 8 + (size_t)K_SEL * 4);

    // 1) scores -> packed keys (WMMA f32 16x16x4). 768 tiles, 8 waves/block.
    score_keys_kernel<<<96, 256, 0, stream>>>(h, W, b, keys);

    // 2) full descending bitonic sort in one WGP's LDS; emit top-K idx/values.
    sort_topk_kernel<<<1, 1024, 0, stream>>>(keys, idx_i32, values, out_idx);

    // 3) new_h: 6144*128 floats = 196608 float4 -> 768 blocks.
    newh_kernel<<<768, 256, 0, stream>>>(h, idx_i32, values, out_newh);

    // 4) sub_g: one block per output row; TDM stages the index table.
    subg_kernel<<<K_SEL, 256, 0, stream>>>(g, idx_i32, out_sub);
}